// graphLeastActionNet_43284680409677
// MI455X (gfx1250) — compile-verified
//
#include <hip/hip_runtime.h>

#define NNODES 100000
#define NEDGES 1600000
#define NCH    32
#define NLAY   4
#define NFIXI  2

typedef __attribute__((ext_vector_type(2))) float v2f;
typedef __attribute__((ext_vector_type(4))) float v4f;
typedef __attribute__((ext_vector_type(8))) float v8f;

__device__ __forceinline__ float silu_f(float x) {
  return x * (1.0f / (1.0f + __expf(-x)));
}

// Exact-f32 WMMA: D = A(16x4) * B(4x16) + C(16x16)
__device__ __forceinline__ v8f wmma4(v2f a, v2f b, v8f c) {
  return __builtin_amdgcn_wmma_f32_16x16x4_f32(false, a, false, b, (short)0, c,
                                               false, false);
}

// ---------------------------------------------------------------------------
// Kernel 1: per 16-node tile (one wave):
//   Z1 = silu([Z_l | f] @ Kf_l)   (16x64 @ 64x32)
//   dZ = Z1 @ K_l                  (16x32 @ 32x32)
// ---------------------------------------------------------------------------
__global__ __launch_bounds__(64)
void node_mm1_kernel(const float* __restrict__ Zl, const float* __restrict__ fin,
                     const float* __restrict__ Kf, const float* __restrict__ Kl,
                     float* __restrict__ dZ)
{
  __shared__ float lA[2][16 * 64];
  __shared__ float lZ1[2][16 * 32];
  const int wave  = threadIdx.x >> 5;
  const int lane  = threadIdx.x & 31;
  const int tile  = blockIdx.x * 2 + wave;
  const int node0 = tile * 16;
  float* A  = lA[wave];
  float* Z1 = lZ1[wave];

  const int mrow = lane & 15;        // A-row / B-C-D column
  const int half = lane >> 4;        // 0 or 1
  const int koff = half * 2;         // K sub-pair offset

  // Stage A = [Z_tile | f_tile] (16 x 64) into LDS, float4 coalesced.
#pragma unroll
  for (int j = 0; j < 8; ++j) {
    int v = j * 32 + lane;           // float4 index 0..255
    int base = v * 4;
    int row = base >> 6;
    int k = base & 63;
    v4f val;
    if (k < 32) val = *(const v4f*)(Zl  + (node0 + row) * NCH + k);
    else        val = *(const v4f*)(fin + (node0 + row) * NCH + (k - 32));
    *(v4f*)(A + base) = val;
  }
  __syncthreads();

  // Z1 = silu(A @ Kf): 2 column blocks x 16 K-steps
#pragma unroll
  for (int nb = 0; nb < 2; ++nb) {
    v8f c = {0.f, 0.f, 0.f, 0.f, 0.f, 0.f, 0.f, 0.f};
#pragma unroll
    for (int s = 0; s < 16; ++s) {
      int k = 4 * s + koff;
      v2f a = *(const v2f*)(A + mrow * 64 + k);
      v2f b;
      b.x = Kf[k * NCH + nb * 16 + mrow];
      b.y = Kf[(k + 1) * NCH + nb * 16 + mrow];
      c = wmma4(a, b, c);
    }
#pragma unroll
    for (int r = 0; r < 8; ++r) {
      int row = r + 8 * half;
      Z1[row * NCH + nb * 16 + mrow] = silu_f(c[r]);
    }
  }
  __syncthreads();

  // dZ = Z1 @ K: 2 column blocks x 8 K-steps
#pragma unroll
  for (int nb = 0; nb < 2; ++nb) {
    v8f c = {0.f, 0.f, 0.f, 0.f, 0.f, 0.f, 0.f, 0.f};
#pragma unroll
    for (int s = 0; s < 8; ++s) {
      int k = 4 * s + koff;
      v2f a = *(const v2f*)(Z1 + mrow * NCH + k);
      v2f b;
      b.x = Kl[k * NCH + nb * 16 + mrow];
      b.y = Kl[(k + 1) * NCH + nb * 16 + mrow];
      c = wmma4(a, b, c);
    }
#pragma unroll
    for (int r = 0; r < 8; ++r) {
      int row = r + 8 * half;
      dZ[(node0 + row) * NCH + nb * 16 + mrow] = c[r];
    }
  }
}

// ---------------------------------------------------------------------------
// Kernel 2: edge phase, one wave per edge (lane = channel):
//   g = silu(ew * (dZ[src] - dZ[dst]));  agg[src] += g (atomic)
// ---------------------------------------------------------------------------
__global__ __launch_bounds__(256)
void edge_kernel(const int* __restrict__ ei0, const int* __restrict__ ei1,
                 const float* __restrict__ ew, const float* __restrict__ dZ,
                 float* __restrict__ agg)
{
  int e = blockIdx.x * 8 + (threadIdx.x >> 5);
  int c = threadIdx.x & 31;
  if (e >= NEDGES) return;
  int s = ei0[e];
  int d = ei1[e];
  float w = ew[e];
  float g = silu_f(w * (dZ[s * NCH + c] - dZ[d * NCH + c]));
  atomicAdd(&agg[s * NCH + c], g);
}

// ---------------------------------------------------------------------------
// Kernel 3: Y_l = -(agg @ K_l^T) (+ X if last layer)
// ---------------------------------------------------------------------------
__global__ __launch_bounds__(64)
void node_mm2_kernel(const float* __restrict__ agg, const float* __restrict__ Kl,
                     const float* __restrict__ X, float* __restrict__ Yl, int addX)
{
  __shared__ float lT[2][16 * 32];
  const int wave  = threadIdx.x >> 5;
  const int lane  = threadIdx.x & 31;
  const int tile  = blockIdx.x * 2 + wave;
  const int node0 = tile * 16;
  float* T = lT[wave];

  const int mrow = lane & 15;
  const int half = lane >> 4;
  const int koff = half * 2;

  // Stage agg tile (16 x 32) into LDS
#pragma unroll
  for (int j = 0; j < 4; ++j) {
    int v = j * 32 + lane;
    int base = v * 4;
    int row = base >> 5;
    int k = base & 31;
    *(v4f*)(T + base) = *(const v4f*)(agg + (node0 + row) * NCH + k);
  }
  __syncthreads();

#pragma unroll
  for (int nb = 0; nb < 2; ++nb) {
    v8f c = {0.f, 0.f, 0.f, 0.f, 0.f, 0.f, 0.f, 0.f};
#pragma unroll
    for (int s = 0; s < 8; ++s) {
      int k = 4 * s + koff;
      v2f a = *(const v2f*)(T + mrow * NCH + k);
      // B = K^T  =>  b[kk][col] = K[col][kk] : contiguous b64 load from K row
      v2f b = *(const v2f*)(Kl + (nb * 16 + mrow) * NCH + k);
      c = wmma4(a, b, c);
    }
#pragma unroll
    for (int r = 0; r < 8; ++r) {
      int row = r + 8 * half;
      int idx = (node0 + row) * NCH + nb * 16 + mrow;
      float y = -c[r];
      if (addX) y += X[idx];
      Yl[idx] = y;
    }
  }
}

// ---------------------------------------------------------------------------
// Kernel 4: closed-form tridiagonal solve over L=4 (elementwise per node-chan)
// ---------------------------------------------------------------------------
__global__ __launch_bounds__(256)
void tridiag_kernel(const float* __restrict__ Y, float* __restrict__ Z, int nc)
{
  int i = blockIdx.x * 256 + threadIdx.x;
  if (i >= nc) return;
  // forward: y_i = a_i*(b_i*y_{i-1} + Y_i),  a_i=sqrt((i+1)/(i+2)), b_i=sqrt(i/(i+1))
  float y0 = 0.70710678f * Y[i];
  float y1 = 0.81649658f * (0.70710678f * y0 + Y[nc + i]);
  float y2 = 0.86602540f * (0.81649658f * y1 + Y[2 * nc + i]);
  float y3 = 0.89442719f * (0.86602540f * y2 + Y[3 * nc + i]);
  // backward: w_{L-1} = sqrt(L/(L+1))*y_{L-1}; w_i = a_i*(a_i*w_{i+1} + y_i)
  float w3 = 0.89442719f * y3;
  float w2 = 0.86602540f * (0.86602540f * w3 + y2);
  float w1 = 0.81649658f * (0.81649658f * w2 + y1);
  float w0 = 0.70710678f * (0.70710678f * w1 + y0);
  Z[i] = w0;
  Z[nc + i] = w1;
  Z[2 * nc + i] = w2;
  Z[3 * nc + i] = w3;
}

__global__ __launch_bounds__(256)
void fill_zero_kernel(float* __restrict__ p, int n)
{
  int i = blockIdx.x * 256 + threadIdx.x;
  if (i < n) p[i] = 0.0f;
}

__global__ __launch_bounds__(256)
void copy_kernel(const float* __restrict__ s, float* __restrict__ d, int n)
{
  int i = blockIdx.x * 256 + threadIdx.x;
  if (i < n) d[i] = s[i];
}

// ---------------------------------------------------------------------------
extern "C" void kernel_launch(void* const* d_in, const int* in_sizes, int n_in,
                              void* d_out, int out_size, void* d_ws, size_t ws_size,
                              hipStream_t stream)
{
  const float* X   = (const float*)d_in[0];   // [N, 32]
  const float* fin = (const float*)d_in[1];   // [N, 32]
  const int*   ei  = (const int*)d_in[2];     // [2, E]
  const float* ew  = (const float*)d_in[3];   // [E]
  const float* K   = (const float*)d_in[4];   // [L, 32, 32]
  const float* Kf  = (const float*)d_in[5];   // [L, 64, 32]
  float* out = (float*)d_out;                 // [N*32] (Z[-1]) ++ [L*N*32] (Z)

  const int NC = NNODES * NCH;                // 3,200,000
  float* Zbuf = out + NC;                     // Z lives in its output slot
  float* dZ   = (float*)d_ws;                 // [N, 32]
  float* agg  = dZ + NC;                      // [N, 32]
  float* Y    = agg + NC;                     // [L, N, 32]

  const int tiles = NNODES / 16;              // 6250
  dim3 nodeGrid(tiles / 2), nodeBlk(64);      // 2 waves/block, 1 tile/wave
  dim3 edgeGrid(NEDGES / 8), edgeBlk(256);    // 8 edges/block, 1 wave/edge
  const int fb = (NC + 255) / 256;

  // Z = 0
  fill_zero_kernel<<<(NC * NLAY + 255) / 256, 256, 0, stream>>>(Zbuf, NC * NLAY);

  for (int it = 0; it < NFIXI; ++it) {
    for (int l = 0; l < NLAY; ++l) {
      fill_zero_kernel<<<fb, 256, 0, stream>>>(agg, NC);
      node_mm1_kernel<<<nodeGrid, nodeBlk, 0, stream>>>(
          Zbuf + l * NC, fin, Kf + l * 2 * NCH * NCH, K + l * NCH * NCH, dZ);
      edge_kernel<<<edgeGrid, edgeBlk, 0, stream>>>(ei, ei + NEDGES, ew, dZ, agg);
      node_mm2_kernel<<<nodeGrid, nodeBlk, 0, stream>>>(
          agg, K + l * NCH * NCH, X, Y + l * NC, (l == NLAY - 1) ? 1 : 0);
    }
    tridiag_kernel<<<fb, 256, 0, stream>>>(Y, Zbuf, NC);
  }

  // out[0:NC] = Z[L-1]
  copy_kernel<<<fb, 256, 0, stream>>>(Zbuf + (NLAY - 1) * NC, out, NC);
}